// NUFFTLayerMultiChannel_59906203845042
// MI455X (gfx1250) — compile-verified
//
#include <hip/hip_runtime.h>
#include <math.h>

// ---------------------------------------------------------------------------
// NUFFT layer, restructured for CDNA5 (gfx1250, wave32, WMMA + TDM).
//   stage 1: windowed Gaussian spreading -> gsumP (K-paired layout), LDS atomics
//   stage 2: frequency filter -> real even circ-conv kernel h (channel-interleaved)
//   stage 3: irfft = gsum (circ-conv) h  as dual-channel f32 WMMA GEMM,
//            h staged into LDS via TENSOR_LOAD_TO_LDS (TDM) when available
//   stage 4: windowed interpolation -> out (B,N,2)
// ---------------------------------------------------------------------------

typedef __attribute__((ext_vector_type(2))) float        v2f;
typedef __attribute__((ext_vector_type(8))) float        v8f;
typedef __attribute__((ext_vector_type(4))) unsigned int v4u;
typedef __attribute__((ext_vector_type(4))) int          v4i;
typedef __attribute__((ext_vector_type(8))) int          v8i;

#define MGRID 2049
#define KPAD  2052            // 513 * 4   (K-dim padded for WMMA K=4 steps)
#define NTILE 129             // ceil(2049/16) output tiles (per channel pair)
#define BATCH 16
#define NPTS  1024
#define HW    38              // Gaussian window halfwidth (exp underflows past ~35.3)

static constexpr double D_PI    = 3.14159265358979323846;
static constexpr double D_TAU   = 2.821e-5;
static constexpr double D_L     = 10.0;
// g = exp(-diff^2/(4 tau)), diff = (2pi/M)(u - m), u = x*M/L
static constexpr double D_ALPHA = (2.0 * D_PI / (double)MGRID) * (2.0 * D_PI / (double)MGRID)
                                  / (4.0 * D_TAU);
static constexpr double D_SCALE = (double)MGRID / D_L;   // x -> grid units

// workspace layout (float offsets)
//   gsumP: paired A layout, gsumP[(k>>1)*32 + b*2 + (k&1)]   size KPAD/2*32 = 32832
//   Hs   : filter (2 x 1025)
//   hker : conv kernel, channel-interleaved hker[m*2 + c]  (2*MGRID floats)
//   irfft: channel-interleaved irfft[b][m][c], size 16*MGRID*2 = 65568
#define OFF_GSUMP 0
#define OFF_HS    32832
#define OFF_HKER  34944
#define OFF_IRFFT 39104

// TDM availability probes (compile-safe on both toolchains)
#if defined(__has_builtin)
#  if __has_builtin(__builtin_amdgcn_tensor_load_to_lds) && \
      __has_builtin(__builtin_amdgcn_s_wait_tensorcnt)
#    define USE_TDM 1
#    if __has_include(<hip/amd_detail/amd_gfx1250_TDM.h>)
#      define TDM_SIX_ARGS 1
#    endif
#  endif
#endif

// ---------------------------------------------------------------------------
// K1: Gaussian spreading. One workgroup per batch row b; accumulate its grid
// row in LDS via ds_add_f32, then scatter into the WMMA-friendly paired layout.
// ---------------------------------------------------------------------------
__global__ __launch_bounds__(256)
void k_spread(const float* __restrict__ x, float* __restrict__ ws) {
    __shared__ float lds_g[MGRID];
    const int b = blockIdx.x;
    float* gsumP = ws + OFF_GSUMP;

    for (int m = threadIdx.x; m < MGRID; m += 256) lds_g[m] = 0.0f;
    __syncthreads();

    for (int n = threadIdx.x; n < NPTS; n += 256) {
        const float u  = x[b * NPTS + n] * (float)D_SCALE;
        const int   c0 = (int)floorf(u);
        const int   lo = max(0, c0 - HW);
        const int   hi = min(MGRID - 1, c0 + HW);
        for (int m = lo; m <= hi; ++m) {
            const float t = u - (float)m;
            atomicAdd(&lds_g[m], expf(-(float)D_ALPHA * t * t));
        }
    }
    __syncthreads();

    for (int m = threadIdx.x; m < MGRID; m += 256)
        gsumP[(m >> 1) * 32 + b * 2 + (m & 1)] = lds_g[m];
    // zero the K-dim padding rows (rows MGRID..KPAD-1, all 16 batch columns)
    if (b == 0)
        for (int i = threadIdx.x; i < (KPAD - MGRID) * BATCH; i += 256) {
            const int m  = MGRID + i / BATCH;
            const int bb = i % BATCH;
            gsumP[(m >> 1) * 32 + bb * 2 + (m & 1)] = 0.0f;
        }
}

// ---------------------------------------------------------------------------
// K2a: Hs[c][k] = mult[c](k) * deconv(k)^2 * (L/2pi) / M,  k = 0..1024.
// Real and even in k -> only non-negative k stored.
// ---------------------------------------------------------------------------
__global__ __launch_bounds__(256)
void k_filter(const float* __restrict__ s0p, const float* __restrict__ a0p,
              const float* __restrict__ s1p, const float* __restrict__ a1p,
              float* __restrict__ ws) {
    const int k = blockIdx.x * 256 + threadIdx.x;
    if (k > MGRID / 2) return;  // 0..1024
    const float s0 = s0p[0], a0 = a0p[0], s1 = s1p[0], a1 = a1p[0];
    const float k2     = (float)k * (float)k;
    const float dec2   = (float)(D_PI / D_TAU) * expf(2.0f * (float)D_TAU * k2);
    const float common = dec2 * (float)(D_L / (2.0 * D_PI)) * (1.0f / (float)MGRID);
    const float d0 = k2 + 25.0f * s0 * s0;
    const float d1 = k2 + 25.0f * s1 * s1;
    const float m1 = -a0 * (float)(4.0 * D_PI) / d0;
    const float m2 =  a1 * (float)(4.0 * D_PI) / (d1 * d1);
    ws[OFF_HS + 0 * 1025 + k] = m1 * common;
    ws[OFF_HS + 1 * 1025 + k] = m2 * common;
}

// ---------------------------------------------------------------------------
// K2b: h[c][m] = Hs[c][0] + 2 * sum_{k=1..1024} Hs[c][k] * cos(2 pi k m / M)
// (real inverse DFT of the even filter). Written channel-interleaved so the
// LDS image in K3 is a flat 1-D tile (TDM-friendly).
// ---------------------------------------------------------------------------
__global__ __launch_bounds__(256)
void k_hker(float* __restrict__ ws) {
    const int gid = blockIdx.x * 256 + threadIdx.x;
    if (gid >= 2 * MGRID) return;
    const int c = gid / MGRID;
    const int m = gid - c * MGRID;
    const float* Hs = ws + OFF_HS + c * 1025;
    const float w = (float)(2.0 * D_PI / (double)MGRID);
    float acc = Hs[0];
    for (int k = 1; k <= MGRID / 2; ++k) {
        const int r = (k * m) % MGRID;             // exact phase reduction (k*m < 2^21)
        acc = fmaf(2.0f * Hs[k], cosf((float)r * w), acc);
    }
    ws[OFF_HKER + m * 2 + c] = acc;                // interleaved [m][c]
}

// ---------------------------------------------------------------------------
// K3: irfft[b][m][c] = sum_k gsum[b][k] * h[c][(m-k) mod M] as f32 WMMA GEMM.
// h (2*MGRID floats, channel-interleaved) is staged into LDS by the Tensor
// Data Mover (TENSOR_LOAD_TO_LDS + s_wait_tensorcnt) when the toolchain
// exposes it; descriptor built per CDNA5 ISA section 8 (1-D tile, 4B elems,
// workgroup_mask=0 since we are not in a cluster).
// A (16x4): rows = batch, one b64 load per lane from the paired gsumP layout
// (a full wave reads one contiguous 256B block). B (4x16): circulant gather
// from channel-interleaved h in LDS, one b64 per index for both channels.
// Two independent accumulator chains (c=0, c=1) share every A fragment.
// ISA 7.12.2 fragment layouts (wave32).
// ---------------------------------------------------------------------------
__global__ __launch_bounds__(256)
void k_conv_wmma(float* __restrict__ ws) {
    __shared__ float h_int[2 * MGRID];             // h_int[m*2 + c]
    const float* __restrict__ gsumP = ws + OFF_GSUMP;
    float* __restrict__ irfft       = ws + OFF_IRFFT;

#if defined(USE_TDM)
    if (threadIdx.x < 32) {                        // wave 0 issues the DMA
        const unsigned long long ga = (unsigned long long)(ws + OFF_HKER);
        const unsigned int lds_addr = (unsigned int)(size_t)&h_int[0];
        // D# group 0: count=1 | lds_addr | global_addr[56:0] | type=2
        v4u g0;
        g0.x = 1u;
        g0.y = lds_addr;
        g0.z = (unsigned int)(ga & 0xffffffffu);
        g0.w = (unsigned int)((ga >> 32) & 0x01ffffffu) | (2u << 30);
        // D# group 1: mask=0, data_size=4B, tensor_dim0=tile_dim0=2*MGRID,
        // tensor_dim1=tile_dim1=1, strides=2*MGRID
        v8i g1;
        g1[0] = (int)(2u << 16);                              // data_size=2 (4B)
        g1[1] = (int)(((2u * MGRID) & 0xffffu) << 16);        // tensor_dim0[15:0]
        g1[2] = (int)((((2u * MGRID) >> 16) & 0xffffu) | (1u << 16)); // dim0 hi | tensor_dim1=1
        g1[3] = (int)(((2u * MGRID) & 0xffffu) << 16);        // tile_dim0
        g1[4] = 1;                                            // tile_dim1=1, tile_dim2=0
        g1[5] = (int)(2u * MGRID);                            // tensor_dim0_stride[31:0]
        g1[6] = (int)(((2u * MGRID) & 0xffffu) << 16);        // stride0 hi=0 | dim1_stride[15:0]
        g1[7] = 0;                                            // dim1_stride[47:16]
        const v4i z4 = {0, 0, 0, 0};
#  if defined(TDM_SIX_ARGS)
        const v8i z8 = {0, 0, 0, 0, 0, 0, 0, 0};
        __builtin_amdgcn_tensor_load_to_lds(g0, g1, z4, z4, z8, 0);
#  else
        __builtin_amdgcn_tensor_load_to_lds(g0, g1, z4, z4, 0);
#  endif
        __builtin_amdgcn_s_wait_tensorcnt(0);
    }
#else
    for (int i = threadIdx.x; i < 2 * MGRID; i += 256)
        h_int[i] = ws[OFF_HKER + i];
#endif
    __syncthreads();

    const int lane = threadIdx.x & 31;
    const int wv   = threadIdx.x >> 5;
    const int col  = lane & 15;          // B column / A batch row
    const int hi2  = (lane >> 4) << 1;   // K-half offset: 0 (lanes 0-15) or 2 (lanes 16-31)
    const int nwaves = gridDim.x * 8;

    for (int t = blockIdx.x * 8 + wv; t < NTILE; t += nwaves) {
        const int m0 = t * 16;

        v8f acc0 = {0.f, 0.f, 0.f, 0.f, 0.f, 0.f, 0.f, 0.f};
        v8f acc1 = {0.f, 0.f, 0.f, 0.f, 0.f, 0.f, 0.f, 0.f};

        // circulant B index for this lane: (m0 + col - (k0 + hi2)) mod M
        int idx = m0 + col - hi2;
        if (idx >= MGRID) idx -= MGRID;
        if (idx < 0)      idx += MGRID;

        for (int k0 = 0; k0 < KPAD; k0 += 4) {
            // A fragment: K pair {k0+hi2, k0+hi2+1} at row=col, contiguous b64
            const v2f a = *(const v2f*)(gsumP + ((k0 + hi2) >> 1) * 32 + col * 2);
            // B fragments: both channels at K=k0+hi2 (idx) and K=k0+hi2+1 (idx-1)
            int j = idx - 1; if (j < 0) j += MGRID;
            const v2f p = *(const v2f*)(h_int + idx * 2);   // {h0[idx], h1[idx]}
            const v2f q = *(const v2f*)(h_int + j   * 2);   // {h0[j],   h1[j]}
            idx -= 4; if (idx < 0) idx += MGRID;
            v2f b0;  b0.x = p.x;  b0.y = q.x;
            v2f b1;  b1.x = p.y;  b1.y = q.y;
#if defined(__has_builtin) && __has_builtin(__builtin_amdgcn_wmma_f32_16x16x4_f32)
            acc0 = __builtin_amdgcn_wmma_f32_16x16x4_f32(
                     false, a, false, b0, (short)0, acc0, false, false);
            acc1 = __builtin_amdgcn_wmma_f32_16x16x4_f32(
                     false, a, false, b1, (short)0, acc1, false, false);
#else
            // wave32 shuffle fallback emulating the WMMA fragment layouts
            #pragma unroll
            for (int r = 0; r < 8; ++r) {
                const int row = r + (hi2 << 2);
                float s0 = 0.0f, s1 = 0.0f;
                #pragma unroll
                for (int k = 0; k < 4; ++k) {
                    const int src = ((k >> 1) << 4);
                    const float Av  = __shfl((k & 1) ? a.y  : a.x,  row + src, 32);
                    const float B0v = __shfl((k & 1) ? b0.y : b0.x, col + src, 32);
                    const float B1v = __shfl((k & 1) ? b1.y : b1.x, col + src, 32);
                    s0 = fmaf(Av, B0v, s0);
                    s1 = fmaf(Av, B1v, s1);
                }
                acc0[r] += s0;
                acc1[r] += s1;
            }
#endif
        }

        // D layout: VGPR r -> row r (lanes 0-15) / row r+8 (lanes 16-31), col = lane%16
        const int m = m0 + col;
        if (m < MGRID) {
            #pragma unroll
            for (int r = 0; r < 8; ++r) {
                const int bb = r + (hi2 << 2);               // batch 0..7 or 8..15
                v2f o;  o.x = acc0[r];  o.y = acc1[r];
                *(v2f*)(irfft + (bb * MGRID + m) * 2) = o;   // interleaved [b][m][c]
            }
        }
    }
}

// ---------------------------------------------------------------------------
// K4: fmm[b][n][c] = (1/M) * sum_m irfft[b][m][c] * g[b][n][m], windowed.
// Channel-interleaved irfft -> one b64 load serves both outputs.
// ---------------------------------------------------------------------------
__global__ __launch_bounds__(256)
void k_interp(const float* __restrict__ x, const float* __restrict__ ws,
              float* __restrict__ out) {
    const int gid = blockIdx.x * 256 + threadIdx.x;
    if (gid >= BATCH * NPTS) return;
    const int b = gid / NPTS;

    const float u  = x[gid] * (float)D_SCALE;
    const int   c0 = (int)floorf(u);
    const int   lo = max(0, c0 - HW);
    const int   hi = min(MGRID - 1, c0 + HW);

    const v2f* __restrict__ rr = (const v2f*)(ws + OFF_IRFFT) + b * MGRID;

    float acc0 = 0.0f, acc1 = 0.0f;
    for (int m = lo; m <= hi; ++m) {
        const float t = u - (float)m;
        const float g = expf(-(float)D_ALPHA * t * t);
        const v2f  v = rr[m];
        acc0 = fmaf(g, v.x, acc0);
        acc1 = fmaf(g, v.y, acc1);
    }
    out[gid * 2 + 0] = acc0 * (1.0f / (float)MGRID);
    out[gid * 2 + 1] = acc1 * (1.0f / (float)MGRID);
}

// ---------------------------------------------------------------------------
extern "C" void kernel_launch(void* const* d_in, const int* in_sizes, int n_in,
                              void* d_out, int out_size, void* d_ws, size_t ws_size,
                              hipStream_t stream) {
    (void)in_sizes; (void)n_in; (void)out_size; (void)ws_size;
    const float* x  = (const float*)d_in[0];
    const float* s0 = (const float*)d_in[1];
    const float* a0 = (const float*)d_in[2];
    const float* s1 = (const float*)d_in[3];
    const float* a1 = (const float*)d_in[4];
    float* ws  = (float*)d_ws;
    float* out = (float*)d_out;

    k_spread   <<<BATCH, 256, 0, stream>>>(x, ws);
    k_filter   <<<(MGRID / 2 + 256) / 256, 256, 0, stream>>>(s0, a0, s1, a1, ws);
    k_hker     <<<(2 * MGRID + 255) / 256, 256, 0, stream>>>(ws);
    k_conv_wmma<<<17, 256, 0, stream>>>(ws);                    // 136 waves >= 129 tiles
    k_interp   <<<(BATCH * NPTS + 255) / 256, 256, 0, stream>>>(x, ws, out);
}